// ClassCaps_88278757802425
// MI455X (gfx1250) — compile-verified
//
#include <hip/hip_runtime.h>

typedef __attribute__((ext_vector_type(2))) float v2f;
typedef __attribute__((ext_vector_type(8))) float v8f;

#define B_   64
#define NI   1152
#define DI   8
#define NO   64
#define DO_  16
#define OD   1024          // NO*DO_
#define NCH  32            // i-chunks for the Ni reduction
#define IPC  36            // NI / NCH

// ---------------------------------------------------------------------------
// GEMM: u_hat[b, i, od] = sum_k W[i, od, k] * x[b, i, k]
// Per block (one i): C[64 x 1024] = X[64 x 8] @ Wt[8 x 1024] via fp32 WMMA.
// A 16x4 f32 frag: lane l -> M = l&15, VGPR j holds K = (l>>4)*2 + j.
// B 4x16 f32 frag: lane l -> N = l&15, VGPR j holds K = (l>>4)*2 + j.
// C/D 16x16 f32 : lane l -> N = l&15, VGPR j holds M = (l>>4)*8 + j.
// ---------------------------------------------------------------------------
__global__ __launch_bounds__(256) void k_gemm(const float* __restrict__ x,
                                              const float* __restrict__ W,
                                              float* __restrict__ uhat)
{
    __shared__ float Xs[64 * 9];     // [b][k], stride 9 to dodge bank conflicts
    __shared__ float Ws[1024 * 9];   // [od][k]
    const int i = blockIdx.x;
    const int t = threadIdx.x;

    for (int j = t; j < 64 * DI; j += 256) {
        int bb = j >> 3, k = j & 7;
        Xs[bb * 9 + k] = x[(size_t)bb * (NI * DI) + (size_t)i * DI + k];
    }
    const float* Wi = W + (size_t)i * (OD * DI);   // contiguous 8192 floats
    for (int j = t; j < OD * DI; j += 256) {
        int od = j >> 3, k = j & 7;
        Ws[od * 9 + k] = Wi[j];
    }
    __syncthreads();

    const int wave = t >> 5;
    const int lane = t & 31;
    const int ml   = lane & 15;
    const int kh   = (lane >> 4) << 1;   // 0 or 2

    for (int mt = 0; mt < 4; ++mt) {
        const int mrow = (mt * 16 + ml) * 9;
        v2f a0, a1;
        a0.x = Xs[mrow + kh + 0];
        a0.y = Xs[mrow + kh + 1];
        a1.x = Xs[mrow + 4 + kh + 0];
        a1.y = Xs[mrow + 4 + kh + 1];
        for (int nt = wave; nt < 64; nt += 8) {
            const int nrow = (nt * 16 + ml) * 9;
            v2f b0, b1;
            b0.x = Ws[nrow + kh + 0];
            b0.y = Ws[nrow + kh + 1];
            b1.x = Ws[nrow + 4 + kh + 0];
            b1.y = Ws[nrow + 4 + kh + 1];
            v8f c = {};
            c = __builtin_amdgcn_wmma_f32_16x16x4_f32(false, a0, false, b0,
                                                      (short)0, c, false, false);
            c = __builtin_amdgcn_wmma_f32_16x16x4_f32(false, a1, false, b1,
                                                      (short)0, c, false, false);
            const int nbase = nt * 16 + ml;
            #pragma unroll
            for (int j = 0; j < 8; ++j) {
                int bb = mt * 16 + (lane >> 4) * 8 + j;
                uhat[(size_t)bb * (NI * OD) + (size_t)i * OD + nbase] = c[j];
            }
        }
    }
}

// ---------------------------------------------------------------------------
// Fused routing step for one iteration. One block per (b, i-chunk).
// MODE 0: c = 1/64 (softmax of zero logits), just weighted partial sum.
// MODE 1: logit = <u_hat, v0>; store logits; c = softmax; partial sum.
// MODE 2: logit = old + <u_hat, v1>; c = softmax -> routing_weights out; partial.
// u_hat row [1024] is read exactly once per (b,i) and reused from registers.
// ---------------------------------------------------------------------------
template<int MODE>
__global__ __launch_bounds__(256) void k_route(const float* __restrict__ uhat,
                                               const float* __restrict__ vin,
                                               const float* __restrict__ logits_in,
                                               float* __restrict__ logits_out,
                                               float* __restrict__ c_out,
                                               float* __restrict__ partial)
{
    const int ch   = blockIdx.x;     // 0..NCH-1
    const int b    = blockIdx.y;     // 0..B_-1
    const int t    = threadIdx.x;    // 0..255
    const int lane = t & 31;

    __shared__ float vs[OD];
    __shared__ float ls[NO];

    if (MODE != 0) {
        #pragma unroll
        for (int j = 0; j < 4; ++j) vs[t + 256 * j] = vin[b * OD + t + 256 * j];
        __syncthreads();
    }

    float acc[4] = {0.f, 0.f, 0.f, 0.f};
    const float* up = uhat + (size_t)b * (NI * OD) + (size_t)(ch * IPC) * OD;

    for (int ii = 0; ii < IPC; ++ii) {
        const int i = ch * IPC + ii;
        float u[4], cc[4];
        #pragma unroll
        for (int j = 0; j < 4; ++j) u[j] = up[t + 256 * j];

        if (MODE == 0) {
            #pragma unroll
            for (int j = 0; j < 4; ++j) cc[j] = 1.0f / 64.0f;
        } else {
            // agreement: dot over Do=16 -> 16-lane butterfly reduction
            float p[4];
            #pragma unroll
            for (int j = 0; j < 4; ++j) p[j] = u[j] * vs[t + 256 * j];
            #pragma unroll
            for (int j = 0; j < 4; ++j) {
                #pragma unroll
                for (int m = 1; m < 16; m <<= 1) p[j] += __shfl_xor(p[j], m, 32);
            }
            if ((t & 15) == 0) {
                #pragma unroll
                for (int j = 0; j < 4; ++j) {
                    const int o = (t >> 4) + 16 * j;
                    float val = p[j];
                    if (MODE == 2)
                        val += logits_in[(size_t)b * (NI * NO) + (size_t)i * NO + o];
                    if (MODE == 1)
                        logits_out[(size_t)b * (NI * NO) + (size_t)i * NO + o] = val;
                    ls[o] = val;
                }
            }
            __syncthreads();
            // softmax over the 64 logits (each wave reduces redundantly)
            const float l0 = ls[lane], l1 = ls[lane + 32];
            float mx = fmaxf(l0, l1);
            #pragma unroll
            for (int m = 1; m < 32; m <<= 1) mx = fmaxf(mx, __shfl_xor(mx, m, 32));
            float ssum = __expf(l0 - mx) + __expf(l1 - mx);
            #pragma unroll
            for (int m = 1; m < 32; m <<= 1) ssum += __shfl_xor(ssum, m, 32);
            const float inv = 1.0f / ssum;
            #pragma unroll
            for (int j = 0; j < 4; ++j)
                cc[j] = __expf(ls[(t >> 4) + 16 * j] - mx) * inv;
            if (MODE == 2 && (t & 15) == 0) {
                #pragma unroll
                for (int j = 0; j < 4; ++j)
                    c_out[(size_t)b * (NI * NO) + (size_t)i * NO + (t >> 4) + 16 * j] = cc[j];
            }
            __syncthreads();   // protect ls before next i
        }
        #pragma unroll
        for (int j = 0; j < 4; ++j) acc[j] += cc[j] * u[j];
        up += OD;
    }

    float* pp = partial + ((size_t)b * NCH + ch) * OD;
    #pragma unroll
    for (int j = 0; j < 4; ++j) pp[t + 256 * j] = acc[j];
}

// ---------------------------------------------------------------------------
// Combine chunk partials -> s[b,o,d], then squash -> v[b,o,d] (optional out).
// One block per b.
// ---------------------------------------------------------------------------
__global__ __launch_bounds__(256) void k_squash(const float* __restrict__ partial,
                                                float* __restrict__ v,
                                                float* __restrict__ out)
{
    const int b = blockIdx.x;
    const int t = threadIdx.x;
    __shared__ float ss[OD];
    __shared__ float sc[NO];

    const float* pp = partial + (size_t)b * NCH * OD;
    #pragma unroll
    for (int j = 0; j < 4; ++j) {
        const int od = t + 256 * j;
        float s = 0.f;
        for (int ch = 0; ch < NCH; ++ch) s += pp[ch * OD + od];
        ss[od] = s;
    }
    __syncthreads();
    if (t < NO) {
        float s2 = 0.f;
        #pragma unroll
        for (int d = 0; d < DO_; ++d) { const float xv = ss[t * DO_ + d]; s2 += xv * xv; }
        sc[t] = s2 / (1.0f + s2) / sqrtf(s2 + 1e-7f);
    }
    __syncthreads();
    #pragma unroll
    for (int j = 0; j < 4; ++j) {
        const int od  = t + 256 * j;
        const float r = sc[od >> 4] * ss[od];
        v[b * OD + od] = r;
        if (out) out[b * OD + od] = r;
    }
}

extern "C" void kernel_launch(void* const* d_in, const int* in_sizes, int n_in,
                              void* d_out, int out_size, void* d_ws, size_t ws_size,
                              hipStream_t stream)
{
    (void)in_sizes; (void)n_in; (void)out_size; (void)ws_size;
    const float* x = (const float*)d_in[0];   // [64,1152,8]
    const float* W = (const float*)d_in[1];   // [1,1152,64,16,8]

    float* out = (float*)d_out;               // [B,No,Do] = 65536 floats
    float* rw  = out + (size_t)B_ * NO * DO_; // [B,Ni,No] routing weights

    float* uhat    = (float*)d_ws;                     // 75,497,472 f
    float* logits  = uhat + (size_t)B_ * NI * OD;      //  4,718,592 f
    float* partial = logits + (size_t)B_ * NI * NO;    //  2,097,152 f
    float* v       = partial + (size_t)B_ * NCH * OD;  //     65,536 f

    k_gemm<<<dim3(NI), dim3(256), 0, stream>>>(x, W, uhat);

    dim3 rg(NCH, B_);
    // it = 0: uniform c
    k_route<0><<<rg, 256, 0, stream>>>(uhat, nullptr, nullptr, nullptr, nullptr, partial);
    k_squash<<<B_, 256, 0, stream>>>(partial, v, nullptr);
    // it = 1: logits = <u_hat, v0>, c = softmax
    k_route<1><<<rg, 256, 0, stream>>>(uhat, v, nullptr, logits, nullptr, partial);
    k_squash<<<B_, 256, 0, stream>>>(partial, v, nullptr);
    // it = 2: logits += <u_hat, v1>, c = softmax -> routing_weights
    k_route<2><<<rg, 256, 0, stream>>>(uhat, v, logits, nullptr, rw, partial);
    k_squash<<<B_, 256, 0, stream>>>(partial, v, out);
}